// Net_41369124995827
// MI455X (gfx1250) — compile-verified
//
#include <hip/hip_runtime.h>
#include <hip/hip_bf16.h>
#include <math.h>

// ---------------- problem constants (from reference) ----------------
constexpr int B    = 128;
constexpr int NPG  = 400;
constexpr int EPG  = 6400;
constexpr int DIM  = 117;           // EMBED == DIM == 117
constexpr int K1   = 320;           // ceil(0.8*400)
constexpr int K2   = 256;           // ceil(0.8*320)
constexpr int N    = B * NPG;       // 51200
constexpr int E    = B * EPG;       // 819200
constexpr int PD   = 128;           // padded feature stride (multiple of 16)
constexpr int LDA  = PD + 4;        // LDS stride to avoid bank conflicts
constexpr int HID  = 51;            // MLP hidden
constexpr int ZD   = 2 * DIM;       // 234

typedef __attribute__((ext_vector_type(2))) float v2f;
typedef __attribute__((ext_vector_type(8))) float v8f;

// ---------------- kernels ----------------

// Zero-pad a 117x117 weight matrix into 128x128.
__global__ void k_padw(const float* __restrict__ w, float* __restrict__ wp) {
    int i = blockIdx.x * blockDim.x + threadIdx.x;
    if (i >= PD * PD) return;
    int r = i >> 7, c = i & (PD - 1);
    wp[i] = (r < DIM && c < DIM) ? w[r * DIM + c] : 0.0f;
}

// h0[i, c] = emb[x[i], c], zero-padded to PD columns.
__global__ void k_embed(const int* __restrict__ x, const float* __restrict__ emb,
                        float* __restrict__ h) {
    int i = blockIdx.x;
    int c = threadIdx.x;
    h[(size_t)i * PD + c] = (c < DIM) ? emb[(size_t)x[i] * DIM + c] : 0.0f;
}

// agg[dst[e], :] += h[src[e], :]   (src < 0 marks an invalidated edge)
__global__ void k_scatter(const float* __restrict__ h, const int* __restrict__ src,
                          const int* __restrict__ dst, float* __restrict__ agg, int ne) {
    const int c = threadIdx.x;          // 0..127 feature column
    const int e0 = blockIdx.x * 16;
    for (int i = 0; i < 16; ++i) {
        int e = e0 + i;
        if (e >= ne) return;
        int s = src[e];
        if (s < 0) continue;
        int d = dst[e];
        float v = h[(size_t)s * PD + c];
        if (v != 0.0f) atomicAdd(&agg[(size_t)d * PD + c], v);
    }
}

// out = relu(A1 @ Bw1 + A2 @ Bw2 + bias), all [M,PD] with PD=128, using
// V_WMMA_F32_16X16X4_F32.  One block per 16-row tile; 8 waves cover the 8
// 16-column tiles; the two A tiles (agg, h) are staged in LDS.
__global__ void k_gemm2_relu(const float* __restrict__ A1, const float* __restrict__ Bw1,
                             const float* __restrict__ A2, const float* __restrict__ Bw2,
                             const float* __restrict__ bias, float* __restrict__ out) {
    __shared__ float lA1[16 * LDA];
    __shared__ float lA2[16 * LDA];
    const int row0 = blockIdx.x * 16;

    for (int i = threadIdx.x; i < 16 * PD; i += blockDim.x) {
        int r = i >> 7, c = i & (PD - 1);
        lA1[r * LDA + c] = A1[(size_t)(row0 + r) * PD + c];
        lA2[r * LDA + c] = A2[(size_t)(row0 + r) * PD + c];
    }
    __syncthreads();

    const int wave = threadIdx.x >> 5;      // 0..7 -> column tile
    const int lane = threadIdx.x & 31;
    const int n0   = wave * 16;
    const int m16  = lane & 15;             // A: row within tile / B: col within tile
    const int kh   = (lane >> 4) * 2;       // K sub-pair selected by lane half

    v8f acc = {};
    for (int k = 0; k < PD; k += 4) {
        const int ka = k + kh;
        v2f a1; a1.x = lA1[m16 * LDA + ka]; a1.y = lA1[m16 * LDA + ka + 1];
        v2f b1; b1.x = Bw1[(size_t)ka * PD + n0 + m16];
                b1.y = Bw1[(size_t)(ka + 1) * PD + n0 + m16];
        acc = __builtin_amdgcn_wmma_f32_16x16x4_f32(false, a1, false, b1,
                                                    (short)0, acc, false, false);
        v2f a2; a2.x = lA2[m16 * LDA + ka]; a2.y = lA2[m16 * LDA + ka + 1];
        v2f b2; b2.x = Bw2[(size_t)ka * PD + n0 + m16];
                b2.y = Bw2[(size_t)(ka + 1) * PD + n0 + m16];
        acc = __builtin_amdgcn_wmma_f32_16x16x4_f32(false, a2, false, b2,
                                                    (short)0, acc, false, false);
    }

    const int col   = n0 + m16;
    const int rbase = row0 + ((lane >> 4) << 3);   // lanes 16..31 hold M = r+8
    const float bv  = (col < DIM) ? bias[col] : 0.0f;
    for (int r = 0; r < 8; ++r) {
        float v = acc[r] + bv;
        v = v > 0.0f ? v : 0.0f;
        if (col >= DIM) v = 0.0f;                  // keep pad columns exactly zero
        out[(size_t)(rbase + r) * PD + col] = v;
    }
}

// score[i] = tanh((h_i . p) / ||p||)
__global__ void k_score(const float* __restrict__ h, const float* __restrict__ p,
                        float* __restrict__ score, int n) {
    __shared__ float sp[DIM];
    __shared__ float spn;
    for (int c = threadIdx.x; c < DIM; c += blockDim.x) sp[c] = p[c];
    __syncthreads();
    if (threadIdx.x == 0) {
        float s = 0.0f;
        for (int c = 0; c < DIM; ++c) s += sp[c] * sp[c];
        spn = sqrtf(s);
    }
    __syncthreads();
    int i = blockIdx.x * blockDim.x + threadIdx.x;
    if (i >= n) return;
    const float* hr = h + (size_t)i * PD;
    float acc = 0.0f;
    for (int c = 0; c < DIM; ++c) acc += hr[c] * sp[c];
    score[i] = tanhf(acc / spn);
}

// Per-graph top-K by rank counting (stable: lower index wins ties, matching
// jax.lax.top_k). Emits old->new map, new->old perm, and kept scores.
__global__ void k_rank(const float* __restrict__ score, int* __restrict__ newidx,
                       int* __restrict__ perm, float* __restrict__ vals,
                       int npg, int K) {
    __shared__ float s[NPG];
    const int g = blockIdx.x;
    for (int i = threadIdx.x; i < npg; i += blockDim.x) s[i] = score[g * npg + i];
    __syncthreads();
    for (int i = threadIdx.x; i < npg; i += blockDim.x) {
        const float si = s[i];
        int rank = 0;
        for (int j = 0; j < npg; ++j) {
            float sj = s[j];
            rank += (sj > si) || (sj == si && j < i);
        }
        if (rank < K) {
            newidx[g * npg + i] = g * K + rank;
            perm[g * K + rank]  = g * npg + i;
            vals[g * K + rank]  = si;
        } else {
            newidx[g * npg + i] = -1;
        }
    }
}

// h_new[i, :] = h[perm[i], :] * vals[i]
__global__ void k_gather(const float* __restrict__ h, const int* __restrict__ perm,
                         const float* __restrict__ vals, float* __restrict__ hn) {
    int i = blockIdx.x;
    int c = threadIdx.x;
    hn[(size_t)i * PD + c] = h[(size_t)perm[i] * PD + c] * vals[i];
}

// Remap edges through newidx; invalid edges become (-1,-1). Safe in-place.
__global__ void k_remap(const int* __restrict__ si, const int* __restrict__ di,
                        const int* __restrict__ newidx, int* __restrict__ so,
                        int* __restrict__ dout, int ne) {
    int e = blockIdx.x * blockDim.x + threadIdx.x;
    if (e >= ne) return;
    int s = si[e], d = di[e];
    int ns = -1, nd = -1;
    if (s >= 0) { ns = newidx[s]; nd = newidx[d]; }
    if (ns < 0 || nd < 0) { so[e] = -1; dout[e] = -1; }
    else                  { so[e] = ns; dout[e] = nd; }
}

// xbuf[g, 0:117] (+)= max over k rows; xbuf[g, 117:234] (+)= mean
__global__ void k_maxmean(const float* __restrict__ h, float* __restrict__ xbuf,
                          int k, int accumulate) {
    int g = blockIdx.x;
    int c = threadIdx.x;
    if (c >= DIM) return;
    const float* base = h + (size_t)g * k * PD + c;
    float m = -INFINITY, sum = 0.0f;
    for (int r = 0; r < k; ++r) {
        float v = base[(size_t)r * PD];
        m = fmaxf(m, v);
        sum += v;
    }
    float mean = sum / (float)k;
    if (accumulate) {
        xbuf[g * ZD + c]       += m;
        xbuf[g * ZD + DIM + c] += mean;
    } else {
        xbuf[g * ZD + c]       = m;
        xbuf[g * ZD + DIM + c] = mean;
    }
}

// out[g] = sigmoid(relu(z_g @ W1 + b1) @ W3 + b3)
__global__ void k_mlp(const float* __restrict__ z, const float* __restrict__ W1,
                      const float* __restrict__ b1, const float* __restrict__ W3,
                      const float* __restrict__ b3, float* __restrict__ out) {
    __shared__ float hz[HID];
    int g = blockIdx.x;
    int j = threadIdx.x;
    if (j < HID) {
        float acc = b1[j];
        const float* zg = z + (size_t)g * ZD;
        for (int i = 0; i < ZD; ++i) acc += zg[i] * W1[i * HID + j];
        hz[j] = acc > 0.0f ? acc : 0.0f;
    }
    __syncthreads();
    if (j == 0) {
        float o = b3[0];
        for (int jj = 0; jj < HID; ++jj) o += hz[jj] * W3[jj];
        out[g] = 1.0f / (1.0f + expf(-o));
    }
}

// ---------------- launcher ----------------
extern "C" void kernel_launch(void* const* d_in, const int* in_sizes, int n_in,
                              void* d_out, int out_size, void* d_ws, size_t ws_size,
                              hipStream_t stream) {
    (void)in_sizes; (void)n_in; (void)out_size; (void)ws_size;

    const int*   x       = (const int*)d_in[0];
    const int*   eidx    = (const int*)d_in[1];
    const int*   src0    = eidx;
    const int*   dst0    = eidx + E;
    const float* emb     = (const float*)d_in[3];
    const float* W_rel1  = (const float*)d_in[4];
    const float* b_rel1  = (const float*)d_in[5];
    const float* W_root1 = (const float*)d_in[6];
    const float* p1      = (const float*)d_in[7];
    const float* W_rel2  = (const float*)d_in[8];
    const float* b_rel2  = (const float*)d_in[9];
    const float* W_root2 = (const float*)d_in[10];
    const float* p2      = (const float*)d_in[11];
    const float* W1      = (const float*)d_in[12];
    const float* b1      = (const float*)d_in[13];
    const float* W3      = (const float*)d_in[14];
    const float* b3      = (const float*)d_in[15];
    float* out = (float*)d_out;

    char* ws = (char*)d_ws;
    size_t off = 0;
    auto alloc = [&](size_t bytes) -> char* {
        char* p = ws + off;
        off = (off + bytes + 255) & ~(size_t)255;
        return p;
    };
    float* H0     = (float*)alloc((size_t)N * PD * 4);   // embeddings / pooled feats
    float* H1     = (float*)alloc((size_t)N * PD * 4);   // conv outputs
    float* AGG    = (float*)alloc((size_t)N * PD * 4);   // scatter accumulator
    int*   SRC2   = (int*)  alloc((size_t)E * 4);
    int*   DST2   = (int*)  alloc((size_t)E * 4);
    int*   NEWIDX = (int*)  alloc((size_t)N * 4);
    int*   PERM   = (int*)  alloc((size_t)B * K1 * 4);
    float* VALS   = (float*)alloc((size_t)B * K1 * 4);
    float* SCORES = (float*)alloc((size_t)N * 4);
    float* WR1    = (float*)alloc((size_t)PD * PD * 4);
    float* WT1    = (float*)alloc((size_t)PD * PD * 4);
    float* WR2    = (float*)alloc((size_t)PD * PD * 4);
    float* WT2    = (float*)alloc((size_t)PD * PD * 4);
    float* XBUF   = (float*)alloc((size_t)B * ZD * 4);

    // --- weight padding + embedding ---
    k_padw<<<(PD * PD + 255) / 256, 256, 0, stream>>>(W_rel1,  WR1);
    k_padw<<<(PD * PD + 255) / 256, 256, 0, stream>>>(W_root1, WT1);
    k_padw<<<(PD * PD + 255) / 256, 256, 0, stream>>>(W_rel2,  WR2);
    k_padw<<<(PD * PD + 255) / 256, 256, 0, stream>>>(W_root2, WT2);
    k_embed<<<N, PD, 0, stream>>>(x, emb, H0);

    // --- conv1: agg = scatter(h0); h1 = relu(agg@Wrel1 + h0@Wroot1 + b1) ---
    hipMemsetAsync(AGG, 0, (size_t)N * PD * 4, stream);
    k_scatter<<<(E + 15) / 16, PD, 0, stream>>>(H0, src0, dst0, AGG, E);
    k_gemm2_relu<<<N / 16, 256, 0, stream>>>(AGG, WR1, H0, WT1, b_rel1, H1);

    // --- pool1 (400 -> 320 per graph) ---
    k_score<<<(N + 127) / 128, 128, 0, stream>>>(H1, p1, SCORES, N);
    k_rank<<<B, 512, 0, stream>>>(SCORES, NEWIDX, PERM, VALS, NPG, K1);
    k_gather<<<B * K1, PD, 0, stream>>>(H1, PERM, VALS, H0);           // H0 <- h1p
    k_remap<<<(E + 255) / 256, 256, 0, stream>>>(src0, dst0, NEWIDX, SRC2, DST2, E);
    k_maxmean<<<B, 128, 0, stream>>>(H0, XBUF, K1, 0);                 // x1

    // --- conv2 on B*K1 nodes ---
    hipMemsetAsync(AGG, 0, (size_t)B * K1 * PD * 4, stream);
    k_scatter<<<(E + 15) / 16, PD, 0, stream>>>(H0, SRC2, DST2, AGG, E);
    k_gemm2_relu<<<(B * K1) / 16, 256, 0, stream>>>(AGG, WR2, H0, WT2, b_rel2, H1);

    // --- pool2 (320 -> 256 per graph) ---
    k_score<<<(B * K1 + 127) / 128, 128, 0, stream>>>(H1, p2, SCORES, B * K1);
    k_rank<<<B, 512, 0, stream>>>(SCORES, NEWIDX, PERM, VALS, K1, K2);
    k_gather<<<B * K2, PD, 0, stream>>>(H1, PERM, VALS, H0);           // H0 <- h2p
    k_maxmean<<<B, 128, 0, stream>>>(H0, XBUF, K2, 1);                 // z = x1 + x2

    // --- MLP head ---
    k_mlp<<<B, 64, 0, stream>>>(XBUF, W1, b1, W3, b3, out);
}